// Stitcher_16527034155146
// MI455X (gfx1250) — compile-verified
//
#include <hip/hip_runtime.h>
#include <stdint.h>

// ---------------------------------------------------------------------------
// Stitcher.merge_dense_tvs on MI455X (gfx1250).
//
// Semantics: out[idx] = value with max |.| among prev[idx] and all client
// values scattered to idx (the sequential per-client scan composes to this).
//
// 32-bit monotonic key:  key(f) = rotl1(bits(f)) = (abs_bits << 1) | sign
//   - ordering by key == ordering by |f|  (ties: negative wins, fixed)
//   - invertible: bits(f) = rotr1(key)
// so global_atomic_max_u32 on a key table implements abs-max-keep-sign.
//
// The 64 MB key table lives *in place* in d_out and is L2-resident (192 MB):
//   1) init:     d_out[i] = key(prev[i])      (b128 NT loads; stores th:WB
//                                              scope:DEV -> dirty in L2, no rinse)
//   2) scatter:  16.8M global_atomic_max_u32 into d_out (random, all L2 hits);
//                inputs streamed with th:NT so they never evict the table
//   3) finalize: d_out[i] = float(rotr1(key)) (loads th:LU -> discard dirty,
//                                              keys never written back to HBM;
//                                              NT b128 store of the result)
// HBM traffic ~320 MB  =>  ~14 us at 23.3 TB/s; atomics run at L2.
// ---------------------------------------------------------------------------

typedef uint32_t u32x4 __attribute__((ext_vector_type(4)));
typedef int32_t  i32x4 __attribute__((ext_vector_type(4)));

static constexpr int kCols    = 4096;
static constexpr int kTotal   = 4096 * 4096;     // 16,777,216 dense elements
static constexpr int kNnz     = 2097152;         // 2^21 per client
static constexpr int kClients = 8;
static constexpr int kEntries = kClients * kNnz; // 16,777,216 updates

__device__ __forceinline__ uint32_t key32(uint32_t b)   { return (b << 1) | (b >> 31); }
__device__ __forceinline__ uint32_t unkey32(uint32_t k) { return (k >> 1) | (k << 31); }

// d_out[i] = key(prev[i]); store kept dirty in L2 (no wr-rinse to HBM).
__global__ void stitch_init(const u32x4* __restrict__ prev,
                            uint32_t*    __restrict__ keys) {
    int i = blockIdx.x * blockDim.x + threadIdx.x;      // one float4 / thread
    u32x4 p = __builtin_nontemporal_load(&prev[i]);
    u32x4 k;
    k.x = key32(p.x); k.y = key32(p.y); k.z = key32(p.z); k.w = key32(p.w);
    uint64_t addr = (uint64_t)(keys + (size_t)i * 4);
    asm volatile("global_store_b128 %0, %1, off th:TH_STORE_WB scope:SCOPE_DEV"
                 :: "v"(addr), "v"(k) : "memory");
}

// One no-return global_atomic_max_u32 per sparse update; table is L2-resident.
__global__ void stitch_scatter(const u32x4* __restrict__ vals,  // [C*NNZ] fp32 bits
                               const int*   __restrict__ idx,   // [C,2,NNZ] int32
                               uint32_t*    __restrict__ keys) {
    int t = blockIdx.x * blockDim.x + threadIdx.x;      // 4 entries / thread
    int e = t << 2;                                     // first entry id
    int c = e >> 21;                                    // kNnz == 2^21
    int j = e & (kNnz - 1);

    u32x4 v = __builtin_nontemporal_load(&vals[t]);
    const int* base = idx + (size_t)c * 2 * kNnz + j;
    i32x4 r  = __builtin_nontemporal_load((const i32x4*)base);          // rows
    i32x4 cc = __builtin_nontemporal_load((const i32x4*)(base + kNnz)); // cols

    atomicMax(&keys[(size_t)(r.x * kCols + cc.x)], key32(v.x));
    atomicMax(&keys[(size_t)(r.y * kCols + cc.y)], key32(v.y));
    atomicMax(&keys[(size_t)(r.z * kCols + cc.z)], key32(v.z));
    atomicMax(&keys[(size_t)(r.w * kCols + cc.w)], key32(v.w));
}

// In-place: d_out[i] = rotr1(d_out[i]).  LU load discards the dirty key line
// after reading it, so the key table never costs HBM writeback bandwidth.
__global__ void stitch_finalize(uint32_t* __restrict__ keys) {
    int i = blockIdx.x * blockDim.x + threadIdx.x;      // one u32x4 / thread
    uint64_t addr = (uint64_t)(keys + (size_t)i * 4);
    u32x4 k;
    asm volatile("global_load_b128 %0, %1, off th:TH_LOAD_LU\n\t"
                 "s_wait_loadcnt 0x0"
                 : "=v"(k) : "v"(addr) : "memory");
    u32x4 o;
    o.x = unkey32(k.x); o.y = unkey32(k.y); o.z = unkey32(k.z); o.w = unkey32(k.w);
    __builtin_nontemporal_store(o, (u32x4*)addr);
}

// ---------------------------------------------------------------------------

extern "C" void kernel_launch(void* const* d_in, const int* in_sizes, int n_in,
                              void* d_out, int out_size, void* d_ws, size_t ws_size,
                              hipStream_t stream) {
    const u32x4* prev_v = (const u32x4*)d_in[0];   // fp32 bits, 4-wide
    const u32x4* vals_v = (const u32x4*)d_in[1];   // fp32 bits, 4-wide
    const int*   idx    = (const int*)d_in[2];     // int32 [C,2,NNZ]
    uint32_t*    out    = (uint32_t*)d_out;        // keys during compute, floats at end

    const dim3 blk(256);                            // 8 wave32 waves / block
    const int dense_blocks  = (kTotal   / 4) / 256; // 16384
    const int update_blocks = (kEntries / 4) / 256; // 16384

    stitch_init    <<<dense_blocks,  blk, 0, stream>>>(prev_v, out);
    stitch_scatter <<<update_blocks, blk, 0, stream>>>(vals_v, idx, out);
    stitch_finalize<<<dense_blocks,  blk, 0, stream>>>(out);
}